// Network_11879879544575
// MI455X (gfx1250) — compile-verified
//
#include <hip/hip_runtime.h>
#include <hip/hip_bf16.h>
#include <math.h>

typedef __attribute__((ext_vector_type(2))) float v2f;
typedef __attribute__((ext_vector_type(8))) float v8f;

#define LEAKK 0.1f

// ---------------------------------------------------------------------------
// Implicit-GEMM 3x3 (dilated) conv using V_WMMA_F32_16X16X4_F32.
// One wave (32 threads) computes a 32(cout) x 32(pixel) tile as a 2x2 grid of
// 16x16 WMMA tiles (4 accumulators). Tensors are [B, Ctot, H, W] with an
// active channel window [c0, c0+cin) / [oc0, oc0+cout) (in-place dense concat).
//
// Operand layouts (ISA 7.12.2, wave32):
//   A (16x4):  lane m=lane&15 holds K={2g,2g+1}, g=lane>>4
//   B (4x16):  lane n=lane&15 holds rows K={2g,2g+1}
//   C/D(16x16): VGPR j -> row j+8g, lane&15 -> column
//
// Main loop: full K-slabs of 4 channels, branch-free, affine pointer
// increments (+36 floats for weights, +4*HW for inputs). Tail slab masks the
// cin%4 remainder on the B operand. Out-of-tile rows/columns are handled by
// address clamping only (they touch only unstored rows/cols of D); spatial
// zero-padding is a mask multiply on B.
// ---------------------------------------------------------------------------
__global__ void __launch_bounds__(32)
conv3x3_wmma(const float* __restrict__ in, int inCtot, int inc0, int cin,
             const float* __restrict__ wgt, const float* __restrict__ bias,
             float* __restrict__ out, int outCtot, int outc0, int cout,
             int Bn, int H, int W, int dil, int doLrelu)
{
    const int lane = threadIdx.x;
    const int HW = H * W;
    const int P  = Bn * HW;
    const int pt = blockIdx.x << 5;            // 32 pixels per block
    const int ct = blockIdx.y << 5;            // 32 couts per block
    const int n  = lane & 15;
    const int g  = lane >> 4;
    const int K9 = cin * 9;

    // ---- two pixel tiles: decode + per-tap offsets/masks ----
    int   offs[2][9];
    float msks[2][9];
    int   pb[2], py[2], px[2];
    bool  pv[2];
    const float* inb[2];
#pragma unroll
    for (int q = 0; q < 2; ++q) {
        const int p  = pt + (q << 4) + n;
        pv[q] = (p < P);
        const int pc = pv[q] ? p : (P - 1);
        const int b  = pc / HW;
        const int r  = pc - b * HW;
        const int y  = r / W;
        const int x  = r - y * W;
        pb[q] = b; py[q] = y; px[q] = x;
        inb[q] = in + ((size_t)b * inCtot + inc0) * HW;
#pragma unroll
        for (int t = 0; t < 9; ++t) {
            const int ky = t / 3;
            const int kx = t - ky * 3;
            const int yy = y + dil * (ky - 1);
            const int xx = x + dil * (kx - 1);
            const bool ok = (yy >= 0) && (yy < H) && (xx >= 0) && (xx < W);
            offs[q][t] = ok ? (yy * W + xx) : 0;
            msks[q][t] = ok ? 1.f : 0.f;
        }
    }

    // ---- two cout tiles: clamped weight row pointers ----
    const float* wrow[2];
#pragma unroll
    for (int a = 0; a < 2; ++a) {
        const int m  = ct + (a << 4) + n;
        const int mc = (m < cout) ? m : (cout - 1);
        wrow[a] = wgt + (size_t)mc * K9;
    }

    v8f acc[2][2];                             // [ctile][ptile]
#pragma unroll
    for (int a = 0; a < 2; ++a)
#pragma unroll
        for (int q = 0; q < 2; ++q)
            acc[a][q] = (v8f){0.f,0.f,0.f,0.f,0.f,0.f,0.f,0.f};

    // ---- main loop: full slabs of 4 channels, affine pointers ----
    const float* ap0 = wrow[0] + (size_t)(g << 1) * 9;
    const float* ap1 = wrow[1] + (size_t)(g << 1) * 9;
    const float* bp0 = inb[0] + (size_t)(g << 1) * HW;
    const float* bp1 = inb[1] + (size_t)(g << 1) * HW;
    const size_t bStep = (size_t)4 * HW;

    const int nfull = cin >> 2;
    for (int it = 0; it < nfull; ++it) {
#pragma unroll
        for (int t = 0; t < 9; ++t) {
            v2f A0; A0.x = ap0[t]; A0.y = ap0[t + 9];
            v2f A1; A1.x = ap1[t]; A1.y = ap1[t + 9];
            v2f B0, B1;
            { const float mk = msks[0][t]; const int o = offs[0][t];
              B0.x = bp0[o] * mk; B0.y = bp0[(size_t)HW + o] * mk; }
            { const float mk = msks[1][t]; const int o = offs[1][t];
              B1.x = bp1[o] * mk; B1.y = bp1[(size_t)HW + o] * mk; }
            acc[0][0] = __builtin_amdgcn_wmma_f32_16x16x4_f32(false, A0, false, B0,
                                                              (short)0, acc[0][0], false, false);
            acc[1][0] = __builtin_amdgcn_wmma_f32_16x16x4_f32(false, A1, false, B0,
                                                              (short)0, acc[1][0], false, false);
            acc[0][1] = __builtin_amdgcn_wmma_f32_16x16x4_f32(false, A0, false, B1,
                                                              (short)0, acc[0][1], false, false);
            acc[1][1] = __builtin_amdgcn_wmma_f32_16x16x4_f32(false, A1, false, B1,
                                                              (short)0, acc[1][1], false, false);
        }
        ap0 += 36; ap1 += 36; bp0 += bStep; bp1 += bStep;
    }

    // ---- tail slab: mask the cin % 4 remainder on B ----
    if (cin & 3) {
        const int base = nfull << 2;
        const int c0 = base + (g << 1), c1 = c0 + 1;
        const float km0 = (c0 < cin) ? 1.f : 0.f;
        const float km1 = (c1 < cin) ? 1.f : 0.f;
        const size_t cc0 = (size_t)((c0 < cin) ? c0 : (cin - 1));
        const size_t cc1 = (size_t)((c1 < cin) ? c1 : (cin - 1));
#pragma unroll
        for (int t = 0; t < 9; ++t) {
            v2f A0; A0.x = wrow[0][cc0 * 9 + t]; A0.y = wrow[0][cc1 * 9 + t];
            v2f A1; A1.x = wrow[1][cc0 * 9 + t]; A1.y = wrow[1][cc1 * 9 + t];
            v2f B0, B1;
            { const float mk = msks[0][t]; const int o = offs[0][t];
              B0.x = inb[0][cc0 * HW + o] * (mk * km0);
              B0.y = inb[0][cc1 * HW + o] * (mk * km1); }
            { const float mk = msks[1][t]; const int o = offs[1][t];
              B1.x = inb[1][cc0 * HW + o] * (mk * km0);
              B1.y = inb[1][cc1 * HW + o] * (mk * km1); }
            acc[0][0] = __builtin_amdgcn_wmma_f32_16x16x4_f32(false, A0, false, B0,
                                                              (short)0, acc[0][0], false, false);
            acc[1][0] = __builtin_amdgcn_wmma_f32_16x16x4_f32(false, A1, false, B0,
                                                              (short)0, acc[1][0], false, false);
            acc[0][1] = __builtin_amdgcn_wmma_f32_16x16x4_f32(false, A0, false, B1,
                                                              (short)0, acc[0][1], false, false);
            acc[1][1] = __builtin_amdgcn_wmma_f32_16x16x4_f32(false, A1, false, B1,
                                                              (short)0, acc[1][1], false, false);
        }
    }

    // ---- stores (guarded) ----
#pragma unroll
    for (int q = 0; q < 2; ++q) {
        if (!pv[q]) continue;
        float* outb = out + ((size_t)pb[q] * outCtot + outc0) * HW
                          + (size_t)py[q] * W + px[q];
#pragma unroll
        for (int a = 0; a < 2; ++a) {
#pragma unroll
            for (int j = 0; j < 8; ++j) {
                const int co = ct + (a << 4) + j + (g << 3);
                if (co < cout) {
                    float v = acc[a][q][j] + bias[co];
                    if (doLrelu) v = (v >= 0.f) ? v : LEAKK * v;
                    outb[(size_t)co * HW] = v;
                }
            }
        }
    }
}

// ---------------------------------------------------------------------------
// Cost volume: out[b, oc0+d, y, x] = lrelu(mean_c f[b,c,y,x]*s[b,c,y+dy-4,x+dx-4])
// ---------------------------------------------------------------------------
__global__ void corr_k(const float* __restrict__ f, int C,
                       const float* __restrict__ s,
                       float* __restrict__ out, int outCtot, int outc0,
                       int Bn, int H, int W)
{
    const int HW = H * W;
    const int P  = Bn * HW;
    const int idx = blockIdx.x * blockDim.x + threadIdx.x;
    if (idx >= 81 * P) return;
    const int d = idx / P;
    const int p = idx - d * P;
    const int b = p / HW;
    const int r = p - b * HW;
    const int y = r / W, x = r - (r / W) * W;
    const int dy = d / 9 - 4, dx = d % 9 - 4;
    const int yy = y + dy, xx = x + dx;
    float sum = 0.f;
    if (yy >= 0 && yy < H && xx >= 0 && xx < W) {
        const float* fp = f + (size_t)b * C * HW + (size_t)y * W + x;
        const float* sp = s + (size_t)b * C * HW + (size_t)yy * W + xx;
        for (int c = 0; c < C; ++c)
            sum += fp[(size_t)c * HW] * sp[(size_t)c * HW];
    }
    float v = sum * (1.0f / (float)C);
    v = (v >= 0.f) ? v : LEAKK * v;
    out[((size_t)b * outCtot + outc0 + d) * HW + (size_t)y * W + x] = v;
}

// ---------------------------------------------------------------------------
// Bilinear warp with zeros padding + 0.999 validity mask (Backward()).
// flow is a standalone [B,2,H,W] tensor; scale multiplies the flow.
// ---------------------------------------------------------------------------
__global__ void warp_k(const float* __restrict__ img, int C,
                       const float* __restrict__ flow, float scale,
                       float* __restrict__ out, int Bn, int H, int W)
{
    const int HW = H * W;
    const int P  = Bn * HW;
    const int p = blockIdx.x * blockDim.x + threadIdx.x;
    if (p >= P) return;
    const int b = p / HW;
    const int r = p - b * HW;
    const int y = r / W, x = r - (r / W) * W;

    const float fx = flow[((size_t)b * 2 + 0) * HW + r];
    const float fy = flow[((size_t)b * 2 + 1) * HW + r];
    const float gx = (float)x + fx * scale;
    const float gy = (float)y + fy * scale;
    const float x0f = floorf(gx), y0f = floorf(gy);
    const float wx = gx - x0f, wy = gy - y0f;
    const int x0 = (int)x0f, y0 = (int)y0f;

    const float m00 = (y0   >= 0 && y0   < H && x0   >= 0 && x0   < W) ? 1.f : 0.f;
    const float m01 = (y0   >= 0 && y0   < H && x0+1 >= 0 && x0+1 < W) ? 1.f : 0.f;
    const float m10 = (y0+1 >= 0 && y0+1 < H && x0   >= 0 && x0   < W) ? 1.f : 0.f;
    const float m11 = (y0+1 >= 0 && y0+1 < H && x0+1 >= 0 && x0+1 < W) ? 1.f : 0.f;
    const float w00 = (1.f - wx) * (1.f - wy), w01 = wx * (1.f - wy);
    const float w10 = (1.f - wx) * wy,         w11 = wx * wy;
    const float mk  = (m00*w00 + m01*w01 + m10*w10 + m11*w11) > 0.999f ? 1.f : 0.f;

    const int xc0 = min(max(x0,     0), W - 1), yc0 = min(max(y0,     0), H - 1);
    const int xc1 = min(max(x0 + 1, 0), W - 1), yc1 = min(max(y0 + 1, 0), H - 1);
    const float* ib = img + (size_t)b * C * HW;
    float* ob = out + (size_t)b * C * HW + r;
    for (int c = 0; c < C; ++c) {
        const float* ic = ib + (size_t)c * HW;
        const float v00 = ic[yc0 * W + xc0] * m00;
        const float v01 = ic[yc0 * W + xc1] * m01;
        const float v10 = ic[yc1 * W + xc0] * m10;
        const float v11 = ic[yc1 * W + xc1] * m11;
        ob[(size_t)c * HW] = (v00*w00 + v01*w01 + v10*w10 + v11*w11) * mk;
    }
}

// ---------------------------------------------------------------------------
// ConvTranspose2d(k=4, s=2, p=1) as lhs-dilated conv (pad=2), cout=2.
// out is a standalone [B,2,2H,2W] tensor.
// ---------------------------------------------------------------------------
__global__ void deconv4x2(const float* __restrict__ in, int inCtot, int inc0, int cin,
                          const float* __restrict__ w, const float* __restrict__ bias,
                          float* __restrict__ out, int Bn, int Hin, int Win)
{
    const int Ho = 2 * Hin, Wo = 2 * Win;
    const int HWo = Ho * Wo, HWi = Hin * Win;
    const int total = Bn * 2 * HWo;
    const int idx = blockIdx.x * blockDim.x + threadIdx.x;
    if (idx >= total) return;
    const int b = idx / (2 * HWo);
    int r = idx - b * 2 * HWo;
    const int co = r / HWo; r -= co * HWo;
    const int Y = r / Wo, X = r - (r / Wo) * Wo;

    float acc = bias[co];
    for (int ky = 0; ky < 4; ++ky) {
        const int iy2 = Y - 2 + ky;
        if (iy2 < 0 || iy2 > 2 * Hin - 2 || (iy2 & 1)) continue;
        const int iy = iy2 >> 1;
        for (int kx = 0; kx < 4; ++kx) {
            const int ix2 = X - 2 + kx;
            if (ix2 < 0 || ix2 > 2 * Win - 2 || (ix2 & 1)) continue;
            const int ix = ix2 >> 1;
            const float* wp = w + (size_t)co * cin * 16 + ky * 4 + kx;
            const float* ip = in + ((size_t)b * inCtot + inc0) * HWi + (size_t)iy * Win + ix;
            float s = 0.f;
            for (int ci = 0; ci < cin; ++ci)
                s += wp[(size_t)ci * 16] * ip[(size_t)ci * HWi];
            acc += s;
        }
    }
    out[idx] = acc;
}

// Copy a channel window between [B,Ctot,H,W] tensors.
__global__ void copyc_k(const float* __restrict__ src, int sCtot, int sc0,
                        float* __restrict__ dst, int dCtot, int dc0,
                        int C, int Bn, int HW)
{
    const int total = Bn * C * HW;
    const int i = blockIdx.x * blockDim.x + threadIdx.x;
    if (i >= total) return;
    const int hw = i % HW;
    const int t = i / HW;
    const int c = t % C;
    const int b = t / C;
    dst[((size_t)b * dCtot + dc0 + c) * HW + hw] =
        src[((size_t)b * sCtot + sc0 + c) * HW + hw];
}

__global__ void add_inplace_k(float* __restrict__ a, const float* __restrict__ b, int n)
{
    const int i = blockIdx.x * blockDim.x + threadIdx.x;
    if (i < n) a[i] += b[i];
}

// Bilinear resize align_corners=True, [4,2,112,256] -> [4,2,448,1024], *20.
__global__ void resize_k(const float* __restrict__ in, float* __restrict__ out)
{
    const int H = 112, W = 256, Ho = 448, Wo = 1024;
    const int total = 4 * 2 * Ho * Wo;
    const int idx = blockIdx.x * blockDim.x + threadIdx.x;
    if (idx >= total) return;
    const int X = idx % Wo;
    int t = idx / Wo;
    const int Y = t % Ho;
    const int bc = t / Ho;
    const float py = (float)Y * (float)(H - 1) / (float)(Ho - 1);
    const float px = (float)X * (float)(W - 1) / (float)(Wo - 1);
    int i0 = (int)floorf(py); if (i0 > H - 2) i0 = H - 2; if (i0 < 0) i0 = 0;
    int j0 = (int)floorf(px); if (j0 > W - 2) j0 = W - 2; if (j0 < 0) j0 = 0;
    const float wy = py - (float)i0, wx = px - (float)j0;
    const float* ib = in + (size_t)bc * H * W;
    const float a00 = ib[i0 * W + j0],       a01 = ib[i0 * W + j0 + 1];
    const float a10 = ib[(i0 + 1) * W + j0], a11 = ib[(i0 + 1) * W + j0 + 1];
    const float v = a00 * (1.f - wy) * (1.f - wx) + a01 * (1.f - wy) * wx
                  + a10 * wy * (1.f - wx)         + a11 * wy * wx;
    out[idx] = 20.f * v;
}

// ---------------------------------------------------------------------------
// Host orchestration
// ---------------------------------------------------------------------------
static inline int cdiv(int a, int b) { return (a + b - 1) / b; }

extern "C" void kernel_launch(void* const* d_in, const int* in_sizes, int n_in,
                              void* d_out, int out_size, void* d_ws, size_t ws_size,
                              hipStream_t stream)
{
    (void)out_size; (void)ws_size;
    const int Bn = 4;
    // level index l = 0..4 <-> pyramid level 2..6
    const int LH_[5] = {112, 56, 28, 14, 7};
    const int LW_[5] = {256, 128, 64, 32, 16};
    const int LC_[5] = {32, 64, 96, 128, 196};
    const int INTC_[5] = {117, 149, 181, 213, 81};

    // ---- canonical sizes of the 100 input leaves ----
    long long sz[100];
    int decBase[5]; int refBase;
    {
        for (int l = 0; l < 5; ++l) {
            sz[l] = (long long)Bn * LC_[l] * LH_[l] * LW_[l];
            sz[5 + l] = sz[l];
        }
        const int couts[6] = {128, 128, 96, 64, 32, 2};
        const int addk[6]  = {0, 128, 256, 352, 416, 448};
        int slot = 10;
        for (int li = 0; li < 5; ++li) {      // dec6, dec5, dec4, dec3, dec2
            const int l = 4 - li;
            decBase[l] = slot;
            for (int i = 0; i < 6; ++i) {
                sz[slot++] = (long long)couts[i] * (INTC_[l] + addk[i]) * 9;
                sz[slot++] = couts[i];
            }
            if (l < 4) {                      // upflow, upfeat
                sz[slot++] = 64;  sz[slot++] = 2;
                sz[slot++] = 2LL * (INTC_[l + 1] + 448) * 16; sz[slot++] = 2;
            }
        }
        refBase = slot;
        const int rco[7] = {128, 128, 128, 96, 64, 32, 2};
        const int rci[7] = {565, 128, 128, 128, 96, 64, 32};
        for (int i = 0; i < 7; ++i) {
            sz[slot++] = (long long)rco[i] * rci[i] * 9;
            sz[slot++] = rco[i];
        }
    }

    // ---- candidate input orderings ----
    static int ord[4][100];
    {
        auto fillIns = [&](int* o, int s) { for (int k = 0; k < 90; ++k) o[s + k] = 10 + k; };
        auto fillSrt = [&](int* o, int s) {
            int p = s;
            for (int l = 0; l < 5; ++l) {                 // dec2..dec6 (sorted keys)
                const int base = decBase[l];
                for (int k = 0; k < 12; ++k) o[p++] = base + k;  // 'convs'
                if (l < 4) {                               // 'upfeat' < 'upflow'
                    o[p++] = base + 14; o[p++] = base + 15;
                    o[p++] = base + 12; o[p++] = base + 13;
                }
            }
            for (int k = 0; k < 14; ++k) o[p++] = refBase + k;
        };
        const int topI[10] = {0, 5, 1, 6, 2, 7, 3, 8, 4, 9};
        { int* o = ord[0]; for (int i = 0; i < 10; ++i) o[i] = topI[i]; fillIns(o, 10); }
        { int* o = ord[1]; for (int i = 0; i < 5; ++i) o[i] = i; fillSrt(o, 5);
          for (int i = 0; i < 5; ++i) o[95 + i] = 5 + i; }
        { int* o = ord[2]; for (int i = 0; i < 10; ++i) o[i] = topI[i]; fillSrt(o, 10); }
        { int* o = ord[3]; for (int i = 0; i < 5; ++i) o[i] = i; fillIns(o, 5);
          for (int i = 0; i < 5; ++i) o[95 + i] = 5 + i; }
    }

    // ---- bind pointers ----
    const float* ptr[100] = {nullptr};
    if (n_in >= 100) {
        bool resolved = false;
        for (int v = 0; v < 4 && !resolved; ++v) {
            bool ok = true;
            for (int i = 0; i < 100; ++i)
                if ((long long)in_sizes[i] != sz[ord[v][i]]) { ok = false; break; }
            if (ok) {
                for (int i = 0; i < 100; ++i) ptr[ord[v][i]] = (const float*)d_in[i];
                resolved = true;
            }
        }
        if (!resolved)
            for (int i = 0; i < 100; ++i) ptr[ord[0][i]] = (const float*)d_in[i];
    } else if (n_in == 11) {
        long long ptot = 0; for (int k = 10; k < 100; ++k) ptot += sz[k];
        int pidx = -1;
        for (int i = 0; i < 11; ++i) if ((long long)in_sizes[i] == ptot) { pidx = i; break; }
        if (pidx < 0) pidx = 10;
        const float* blob = (const float*)d_in[pidx];
        long long off = 0;
        for (int k = 10; k < 100; ++k) { ptr[k] = blob + off; off += sz[k]; }
        if (pidx == 5) {
            for (int i = 0; i < 5; ++i) { ptr[i] = (const float*)d_in[i]; ptr[5 + i] = (const float*)d_in[6 + i]; }
        } else {
            const int topI[10] = {0, 5, 1, 6, 2, 7, 3, 8, 4, 9};
            int j = 0;
            for (int i = 0; i < 11; ++i) { if (i == pidx) continue; ptr[topI[j++]] = (const float*)d_in[i]; }
        }
    } else {
        for (int i = 0; i < n_in && i < 100; ++i) ptr[ord[0][i]] = (const float*)d_in[i];
    }

    struct CP { const float* w; const float* b; };
    CP decConv[5][6], decUpflow[4], decUpfeat[4], refc[7];
    for (int l = 0; l < 5; ++l) {
        const int base = decBase[l];
        for (int i = 0; i < 6; ++i) { decConv[l][i].w = ptr[base + 2*i]; decConv[l][i].b = ptr[base + 2*i + 1]; }
        if (l < 4) {
            decUpflow[l].w = ptr[base + 12]; decUpflow[l].b = ptr[base + 13];
            decUpfeat[l].w = ptr[base + 14]; decUpfeat[l].b = ptr[base + 15];
        }
    }
    for (int i = 0; i < 7; ++i) { refc[i].w = ptr[refBase + 2*i]; refc[i].b = ptr[refBase + 2*i + 1]; }
    const float* F[5]; const float* S[5];
    for (int l = 0; l < 5; ++l) { F[l] = ptr[l]; S[l] = ptr[5 + l]; }

    // ---- workspace layout (floats) ----
    float* ws = (float*)d_ws;
    size_t nPix[5];
    for (int l = 0; l < 5; ++l) nPix[l] = (size_t)Bn * LH_[l] * LW_[l];
    size_t featOff[5], flowOff[5], upflowOff[4], upfeatOff[4], warpOff[4], refinedOff;
    {
        size_t off = 0;
        for (int l = 0; l < 5; ++l) { featOff[l] = off; off += (size_t)(INTC_[l] + 448) * nPix[l]; }
        for (int l = 0; l < 5; ++l) { flowOff[l] = off; off += 2 * nPix[l]; }
        for (int l = 0; l < 4; ++l) {
            upflowOff[l] = off; off += 2 * nPix[l];
            upfeatOff[l] = off; off += 2 * nPix[l];
        }
        for (int l = 0; l < 4; ++l) { warpOff[l] = off; off += (size_t)LC_[l] * nPix[l]; }
        refinedOff = off; off += 2 * nPix[0];
    }

    auto launch_conv = [&](const float* in, int iCt, int ic0, int cin, CP p,
                           float* out, int oCt, int oc0, int cout,
                           int H, int W, int dil, int act) {
        dim3 grid((unsigned)cdiv(Bn * H * W, 32), (unsigned)cdiv(cout, 32), 1);
        conv3x3_wmma<<<grid, dim3(32, 1, 1), 0, stream>>>(
            in, iCt, ic0, cin, p.w, p.b, out, oCt, oc0, cout, Bn, H, W, dil, act);
    };

    auto runChain = [&](int l) {
        const int H = LH_[l], W = LW_[l];
        const int ci = INTC_[l], Cm = ci + 448;
        float* feat = ws + featOff[l];
        float* flw  = ws + flowOff[l];
        const int couts[6] = {128, 128, 96, 64, 32, 2};
        const int addk[6]  = {0, 128, 256, 352, 416, 448};
        const int inoff[6] = {448, 320, 192, 96, 32, 0};
        const int outoff[5] = {320, 192, 96, 32, 0};
        for (int i = 0; i < 5; ++i)
            launch_conv(feat, Cm, inoff[i], ci + addk[i], decConv[l][i],
                        feat, Cm, outoff[i], couts[i], H, W, 1, 1);
        launch_conv(feat, Cm, 0, ci + 448, decConv[l][5], flw, 2, 0, 2, H, W, 1, 0);
    };

    // ---- decoder level 6 ----
    {
        const int l = 4, H = LH_[4], W = LW_[4], C = LC_[4];
        const int nP = (int)nPix[4];
        const int Cm = INTC_[4] + 448;
        float* feat = ws + featOff[4];
        corr_k<<<cdiv(81 * nP, 256), 256, 0, stream>>>(F[l], C, S[l], feat, Cm, 448, Bn, H, W);
        runChain(4);
    }

    // ---- decoder levels 5..2 ----
    const float dblv[4] = {5.0f, 2.5f, 1.25f, 0.625f};   // for levels 2,3,4,5
    for (int l = 3; l >= 0; --l) {
        const int u = l + 1;
        const int H = LH_[l], W = LW_[l], C = LC_[l];
        const int HW = H * W, nP = (int)nPix[l];
        const int Hu = LH_[u], Wu = LW_[u];
        const int Cm = INTC_[l] + 448;
        const int CmU = INTC_[u] + 448;
        float* feat  = ws + featOff[l];
        float* featU = ws + featOff[u];
        float* flowU = ws + flowOff[u];
        float* upfl  = ws + upflowOff[l];
        float* upft  = ws + upfeatOff[l];
        float* wbuf  = ws + warpOff[l];

        const int dtot = Bn * 2 * H * W;
        deconv4x2<<<cdiv(dtot, 256), 256, 0, stream>>>(flowU, 2, 0, 2,
            decUpflow[l].w, decUpflow[l].b, upfl, Bn, Hu, Wu);
        deconv4x2<<<cdiv(dtot, 256), 256, 0, stream>>>(featU, CmU, 0, CmU,
            decUpfeat[l].w, decUpfeat[l].b, upft, Bn, Hu, Wu);
        warp_k<<<cdiv(nP, 256), 256, 0, stream>>>(S[l], C, upfl, dblv[l], wbuf, Bn, H, W);
        corr_k<<<cdiv(81 * nP, 256), 256, 0, stream>>>(F[l], C, wbuf, feat, Cm, 448, Bn, H, W);
        copyc_k<<<cdiv(C * nP, 256), 256, 0, stream>>>(F[l], C, 0, feat, Cm, 529, C, Bn, HW);
        copyc_k<<<cdiv(2 * nP, 256), 256, 0, stream>>>(upfl, 2, 0, feat, Cm, 529 + C, 2, Bn, HW);
        copyc_k<<<cdiv(2 * nP, 256), 256, 0, stream>>>(upft, 2, 0, feat, Cm, 531 + C, 2, Bn, HW);
        runChain(l);
    }

    // ---- refiner (ping-pong through dead feat3/feat2 regions) ----
    {
        const int H = 112, W = 256;
        const int nP = (int)nPix[0];
        float* refA = ws + featOff[1];    // feat3 region (>= 128*nP floats)
        float* refB = ws + featOff[0];    // feat2 region (dead after conv0)
        float* refined = ws + refinedOff;
        const int rco[7] = {128, 128, 128, 96, 64, 32, 2};
        const int rci[7] = {565, 128, 128, 128, 96, 64, 32};
        const int rdil[7] = {1, 2, 4, 8, 16, 1, 1};

        launch_conv(ws + featOff[0], 565, 0, 565, refc[0], refA, rco[0], 0, rco[0], H, W, rdil[0], 1);
        const float* cur = refA;
        for (int i = 1; i < 6; ++i) {
            float* nxt = (cur == refA) ? refB : refA;
            launch_conv(cur, rci[i], 0, rci[i], refc[i], nxt, rco[i], 0, rco[i], H, W, rdil[i], 1);
            cur = nxt;
        }
        launch_conv(cur, rci[6], 0, rci[6], refc[6], refined, 2, 0, 2, H, W, rdil[6], 0);

        add_inplace_k<<<cdiv(2 * nP, 256), 256, 0, stream>>>(refined, ws + flowOff[0], 2 * nP);

        const int total = 4 * 2 * 448 * 1024;
        resize_k<<<cdiv(total, 256), 256, 0, stream>>>(refined, (float*)d_out);
    }
}